// HierarchicalClustering_2671469658719
// MI455X (gfx1250) — compile-verified
//
#include <hip/hip_runtime.h>

// CDNA5 / gfx1250 Ward agglomerative clustering.
// Phase 1: d2 = sq_i + sq_j - 2*X*X^T via V_WMMA_F32_16X16X4_F32 (f32 precision),
//          D=96 compile-time -> fully unrolled 24-chunk WMMA chain.
// Phase 2: persistent single-workgroup merge loop (n-k sequential merges),
//          rowmin nearest-neighbor cache in LDS, d2 resident in 192MB L2.

typedef __attribute__((ext_vector_type(2))) float v2f;
typedef __attribute__((ext_vector_type(8))) float v8f;

#define MERGE_THREADS 1024
#define MAX_N 2048
#define FEAT_D 96           // feature dim fixed by reference shape [4,512,96]

// ---------------------------------------------------------------- init
__global__ void ward_init(const float* __restrict__ x, float* __restrict__ sq,
                          int* __restrict__ parent, int n) {
    int i = blockIdx.x * blockDim.x + threadIdx.x;
    if (i >= n) return;
    const float* row = x + (size_t)i * FEAT_D;
    float s = 0.0f;
#pragma unroll 8
    for (int c = 0; c < FEAT_D; ++c) s += row[c] * row[c];
    sq[i] = s;
    parent[i] = i;
}

// ------------------------------------------------- WMMA distance matrix
// One wave (32 lanes) per 16x16 tile of d2.  A: 16x4 f32 (2 VGPR/lane),
// B: 4x16 f32 (2 VGPR/lane), C/D: 16x16 f32 (8 VGPR/lane).
// K = 96 -> 24 fully unrolled v_wmma_f32_16x16x4_f32.
__global__ __launch_bounds__(32)
void ward_d2_wmma(const float* __restrict__ x, const float* __restrict__ sq,
                  float* __restrict__ d2, int n) {
    const int l = threadIdx.x & 15;   // lane-in-half
    const int h = threadIdx.x >> 4;   // half select (K pairs {0,1} vs {2,3})
    const int i0 = blockIdx.y * 16;
    const int j0 = blockIdx.x * 16;
    const float* arow = x + (size_t)(i0 + l) * FEAT_D + 2 * h;
    const float* brow = x + (size_t)(j0 + l) * FEAT_D + 2 * h;

    v8f c = {};
#pragma unroll
    for (int k0 = 0; k0 < FEAT_D; k0 += 4) {
        v2f a, b;
        a.x = arow[k0];     a.y = arow[k0 + 1];
        b.x = brow[k0];     b.y = brow[k0 + 1];
        // D = A(16x4) * B(4x16) + C, f32 accumulate
        c = __builtin_amdgcn_wmma_f32_16x16x4_f32(
                /*neg_a=*/false, a, /*neg_b=*/false, b,
                /*c_mod=*/(short)0, c, /*reuse_a=*/false, /*reuse_b=*/false);
    }

    const float INF = __builtin_inff();
    const float sqc = sq[j0 + l];     // column norm: same for all 8 rows this lane owns
#pragma unroll
    for (int r = 0; r < 8; ++r) {
        int row = i0 + r + 8 * h;   // C/D layout: VGPR r -> M=r (lanes 0-15), M=r+8 (16-31)
        int col = j0 + l;
        float v = sq[row] + sqc - 2.0f * c[r];
        v = fmaxf(v, 0.0f);
        if (row == col) v = INF;
        d2[(size_t)row * n + col] = v;
    }
}

// ------------------------------------------------------ persistent merge
__global__ __launch_bounds__(MERGE_THREADS)
void ward_merge(float* __restrict__ d2, int* __restrict__ parent,
                int* __restrict__ activeOut, int n, int k) {
    __shared__ float rowmin[MAX_N];
    __shared__ int   rowarg[MAX_N];
    __shared__ float sizesL[MAX_N];
    __shared__ int   activeL[MAX_N];
    __shared__ float redv[MERGE_THREADS];
    __shared__ int   redi[MERGE_THREADS];
    __shared__ int   flagList[MAX_N];
    __shared__ int   flagCnt;
    __shared__ int   sI, sJ;
    __shared__ float sDij, sSi, sSj;

    const int t = threadIdx.x;
    const int lane = t & 31;
    const int wave = t >> 5;
    const int nwaves = MERGE_THREADS / 32;
    const float INF = __builtin_inff();

    for (int m = t; m < n; m += MERGE_THREADS) { sizesL[m] = 1.0f; activeL[m] = 1; }
    __syncthreads();

    // initial per-row nearest neighbor: one wave per row (coalesced scan)
    for (int r = wave; r < n; r += nwaves) {
        float v = INF; int a = -1;
        const float* row = d2 + (size_t)r * n;
        for (int cc = lane; cc < n; cc += 32) {
            float q = row[cc];
            if (q < v) { v = q; a = cc; }
        }
        for (int off = 16; off > 0; off >>= 1) {
            float ov = __shfl_xor(v, off, 32);
            int   oa = __shfl_xor(a, off, 32);
            if (ov < v) { v = ov; a = oa; }
        }
        if (lane == 0) { rowmin[r] = v; rowarg[r] = a; }
    }
    __syncthreads();

    const int merges = n - k;
    for (int it = 0; it < merges; ++it) {
        // ---- A: global argmin over rowmin (block tree reduction in LDS)
        float v = INF; int a = -1;
        for (int m = t; m < n; m += MERGE_THREADS) {
            float q = rowmin[m];
            if (q < v) { v = q; a = m; }
        }
        redv[t] = v; redi[t] = a;
        __syncthreads();
        for (int s = MERGE_THREADS / 2; s > 0; s >>= 1) {
            if (t < s && redv[t + s] < redv[t]) { redv[t] = redv[t + s]; redi[t] = redi[t + s]; }
            __syncthreads();
        }

        // ---- B: resolve merge pair, broadcast scalars
        if (t == 0) {
            int i = redi[0];
            int j = rowarg[i];
            if (i > j) { int tmp = i; i = j; j = tmp; }
            sI = i; sJ = j; sDij = redv[0];
            sSi = sizesL[i]; sSj = sizesL[j];
            flagCnt = 0;
        }
        __syncthreads();
        const int   i = sI,  j = sJ;
        const float dij = sDij, si = sSi, sj = sSj;

        // ---- C: Lance-Williams Ward update over active clusters m
        for (int m = t; m < n; m += MERGE_THREADS) {
            if (m == i || m == j || !activeL[m]) continue;
            float sm  = sizesL[m];
            float dim = d2[(size_t)i * n + m];
            float djm = d2[(size_t)j * n + m];
            float nd = ((si + sm) * dim + (sj + sm) * djm - sm * dij) / (si + sj + sm);
            d2[(size_t)i * n + m] = nd;
            d2[(size_t)m * n + i] = nd;
            d2[(size_t)j * n + m] = INF;
            d2[(size_t)m * n + j] = INF;
            float old = rowmin[m];
            if (nd < old) { rowmin[m] = nd; rowarg[m] = i; }
            else {
                int ra = rowarg[m];
                if (ra == i || ra == j) {       // cached NN invalidated -> recompute
                    int slot = atomicAdd(&flagCnt, 1);
                    flagList[slot] = m;
                }
            }
        }
        if (t == 0) {
            d2[(size_t)i * n + j] = INF;
            d2[(size_t)j * n + i] = INF;
            parent[j] = i;
            sizesL[i] = si + sj;
            activeL[j] = 0;
            rowmin[j] = INF;
            int slot = atomicAdd(&flagCnt, 1);  // row i always needs recompute
            flagList[slot] = i;
        }
        __threadfence_block();
        __syncthreads();

        // ---- D: recompute invalidated rows (one wave per flagged row)
        int fc = flagCnt;
        for (int f = wave; f < fc; f += nwaves) {
            int r = flagList[f];
            float rv = INF; int ra = -1;
            const float* row = d2 + (size_t)r * n;
            for (int cc = lane; cc < n; cc += 32) {
                float q = row[cc];
                if (q < rv) { rv = q; ra = cc; }
            }
            for (int off = 16; off > 0; off >>= 1) {
                float ov = __shfl_xor(rv, off, 32);
                int   oa = __shfl_xor(ra, off, 32);
                if (ov < rv) { rv = ov; ra = oa; }
            }
            if (lane == 0) { rowmin[r] = rv; rowarg[r] = ra; }
        }
        __syncthreads();
    }

    for (int m = t; m < n; m += MERGE_THREADS) activeOut[m] = activeL[m];
}

// ----------------------------------------------------------- relabel
// label(p) = rank of root(p) among sorted surviving cluster ids
// (matches np.unique(cluster_of, return_inverse=True))
__global__ void ward_label(const int* __restrict__ parent, const int* __restrict__ active,
                           int* __restrict__ labels, int n) {
    int p = blockIdx.x * blockDim.x + threadIdx.x;
    if (p >= n) return;
    int r = p;
    while (parent[r] != r) r = parent[r];   // chains strictly decrease -> terminates
    int cnt = 0;
    for (int c = 0; c < r; ++c) cnt += (active[c] != 0);
    labels[p] = cnt;
}

// ----------------------------------------------------------- one-hot
__global__ void ward_onehot(const int* __restrict__ labels, float* __restrict__ out,
                            int n, int k) {
    int idx = blockIdx.x * blockDim.x + threadIdx.x;
    if (idx >= n * k) return;
    int p = idx / k;
    int q = idx - p * k;
    out[idx] = (labels[p] == q) ? 1.0f : 0.0f;
}

// ----------------------------------------------------------- launcher
extern "C" void kernel_launch(void* const* d_in, const int* in_sizes, int n_in,
                              void* d_out, int out_size, void* d_ws, size_t ws_size,
                              hipStream_t stream) {
    const float* x = (const float*)d_in[0];
    const int n = in_sizes[0] / FEAT_D;  // 2048 points
    const int k = out_size / n;          // 100 clusters

    // workspace layout: d2 (n*n f32 = 16MB), sq, parent, active, labels
    float* d2     = (float*)d_ws;
    float* sq     = d2 + (size_t)n * n;
    int*   parent = (int*)(sq + n);
    int*   active = parent + n;
    int*   labels = active + n;

    ward_init<<<(n + 255) / 256, 256, 0, stream>>>(x, sq, parent, n);

    dim3 grid(n / 16, n / 16);
    ward_d2_wmma<<<grid, 32, 0, stream>>>(x, sq, d2, n);

    ward_merge<<<1, MERGE_THREADS, 0, stream>>>(d2, parent, active, n, k);

    ward_label<<<(n + 255) / 256, 256, 0, stream>>>(parent, active, labels, n);

    ward_onehot<<<((n * k) + 255) / 256, 256, 0, stream>>>(labels, (float*)d_out, n, k);
}